// CPABWarp_20512763805840
// MI455X (gfx1250) — compile-verified
//
#include <hip/hip_runtime.h>
#include <hip/hip_bf16.h>

typedef __attribute__((ext_vector_type(2))) float v2f;
typedef __attribute__((ext_vector_type(8))) float v8f;

#define BSZ     64
#define SEQ     4096
#define DIM     128
#define DTH     63      // d_theta = N_CELLS - 1
#define NCELLS  64
#define NSTEPS  50
#define NCHUNK  16      // S split for pooling

// ---------------------------------------------------------------------------
// 1) Pooling: partial[chunk][b][d] = sum over 256 rows of input_seq[b, :, d]
//    float4 (b128) fully-coalesced streaming loads; LDS reduce. Deterministic.
// ---------------------------------------------------------------------------
__global__ void pool_kernel(const float* __restrict__ in, float* __restrict__ partial) {
    __shared__ float4 sdata[256];
    const int b     = blockIdx.x >> 4;        // / NCHUNK
    const int chunk = blockIdx.x & (NCHUNK - 1);
    const int q  = threadIdx.x & 31;          // which float4 of the 128-wide row
    const int sg = threadIdx.x >> 5;          // 0..7 row-group
    const float4* base =
        (const float4*)(in + ((size_t)b * SEQ + (size_t)chunk * 256) * DIM);
    float4 sum = make_float4(0.f, 0.f, 0.f, 0.f);
    #pragma unroll 4
    for (int j = 0; j < 32; ++j) {
        float4 v = base[(size_t)(8 * j + sg) * 32 + q];
        sum.x += v.x; sum.y += v.y; sum.z += v.z; sum.w += v.w;
    }
    sdata[threadIdx.x] = sum;
    __syncthreads();
    if (threadIdx.x < 32) {
        float4 t = sdata[threadIdx.x];
        #pragma unroll
        for (int g = 1; g < 8; ++g) {
            float4 u = sdata[g * 32 + threadIdx.x];
            t.x += u.x; t.y += u.y; t.z += u.z; t.w += u.w;
        }
        ((float4*)(partial + (size_t)chunk * (BSZ * DIM) + (size_t)b * DIM))[threadIdx.x] = t;
    }
}

// ---------------------------------------------------------------------------
// 2) Fold chunk partials -> pooled mean (64 x 128)
// ---------------------------------------------------------------------------
__global__ void reduce_kernel(const float* __restrict__ partial, float* __restrict__ pooled) {
    const int i = blockIdx.x * 256 + threadIdx.x;   // 8192 total
    float s = 0.f;
    #pragma unroll
    for (int c = 0; c < NCHUNK; ++c) s += partial[c * (BSZ * DIM) + i];
    pooled[i] = s * (1.0f / (float)SEQ);
}

// ---------------------------------------------------------------------------
// 3a) cvec[j] = sum_k b_loc[k] * basis[j][k]   (b_loc @ basis^T)
// ---------------------------------------------------------------------------
__global__ void cvec_kernel(const float* __restrict__ b_loc, const float* __restrict__ basis,
                            float* __restrict__ cvec) {
    const int j = threadIdx.x;                      // 0..127
    float s = 0.f;
    for (int k = 0; k < DTH; ++k) s += b_loc[k] * basis[j * DTH + k];
    cvec[j] = s;
}

// ---------------------------------------------------------------------------
// 3b) Mmat (128x128) = W_loc (128x63) @ basis^T (63x128), K padded to 64.
//     64 WMMA tiles (8x8), one wave per tile, 16x v_wmma_f32_16x16x4_f32.
//     Fragment layouts per ISA 7.12.2: element K index = 2*(lane/16) + vgpr.
// ---------------------------------------------------------------------------
__global__ void prep_M_kernel(const float* __restrict__ W_loc, const float* __restrict__ basis,
                              float* __restrict__ Mmat) {
    const int lane = threadIdx.x & 31;
    const int wave = threadIdx.x >> 5;
    const int tile = blockIdx.x * 8 + wave;         // 0..63
    const int ti = tile >> 3;                       // row tile (i over D=128)
    const int tj = tile & 7;                        // col tile (j over 128)
    const int m  = lane & 15;
    const int hi = lane >> 4;
    const int row = ti * 16 + m;                    // A-matrix M index
    const int col = tj * 16 + m;                    // B-matrix N index
    v8f acc = {};
    for (int k0 = 0; k0 < 64; k0 += 4) {
        v2f a, b;
        #pragma unroll
        for (int v = 0; v < 2; ++v) {
            const int k = k0 + 2 * hi + v;
            a[v] = (k < DTH) ? W_loc[row * DTH + k] : 0.0f;  // W_loc[i][k]
            b[v] = (k < DTH) ? basis[col * DTH + k] : 0.0f;  // basis^T[k][j] = basis[j][k]
        }
        acc = __builtin_amdgcn_wmma_f32_16x16x4_f32(
            false, a, false, b, (short)0, acc, false, false);
    }
    #pragma unroll
    for (int v = 0; v < 8; ++v) {
        const int r = ti * 16 + v + 8 * hi;
        const int c = tj * 16 + (lane & 15);
        Mmat[r * 128 + c] = acc[v];
    }
}

// ---------------------------------------------------------------------------
// 3c) Acoef (64x128) = pooled (64x128) @ Mmat (128x128) + cvec
//     32 WMMA tiles (4x8), one wave per tile, 32x v_wmma_f32_16x16x4_f32.
// ---------------------------------------------------------------------------
__global__ void gemm_A_kernel(const float* __restrict__ pooled, const float* __restrict__ Mmat,
                              const float* __restrict__ cvec, float* __restrict__ Acoef) {
    const int lane = threadIdx.x & 31;
    const int wave = threadIdx.x >> 5;
    const int tile = blockIdx.x * 8 + wave;         // 0..31
    const int ti = tile >> 3;                       // row tile (b over 64)
    const int tj = tile & 7;                        // col tile (over 128)
    const int m  = lane & 15;
    const int hi = lane >> 4;
    v8f acc = {};
    for (int k0 = 0; k0 < 128; k0 += 4) {
        v2f a, b;
        #pragma unroll
        for (int v = 0; v < 2; ++v) {
            const int k = k0 + 2 * hi + v;
            a[v] = pooled[(ti * 16 + m) * DIM + k];     // A: rows = batch
            b[v] = Mmat[k * 128 + tj * 16 + m];         // B: rows = k
        }
        acc = __builtin_amdgcn_wmma_f32_16x16x4_f32(
            false, a, false, b, (short)0, acc, false, false);
    }
    #pragma unroll
    for (int v = 0; v < 8; ++v) {
        const int r = ti * 16 + v + 8 * hi;
        const int c = tj * 16 + (lane & 15);
        Acoef[r * 128 + c] = acc[v] + cvec[c];
    }
}

// ---------------------------------------------------------------------------
// 4) CPAB ODE integration: one thread per (b, s) point. Coefficient row
//    (a_c,b_c interleaved, 128 f32) cached in LDS; (a,b) gather = ds_load_b64.
// ---------------------------------------------------------------------------
__global__ void ode_kernel(const float* __restrict__ Acoef, float* __restrict__ out) {
    __shared__ float coef[2 * NCELLS];
    const int b     = blockIdx.x >> 4;              // S/256 = 16 chunks per batch
    const int chunk = blockIdx.x & 15;
    if (threadIdx.x < 128) coef[threadIdx.x] = Acoef[b * 128 + threadIdx.x];
    __syncthreads();
    const float2* cp = (const float2*)coef;         // cp[c] = (a_c, b_c), 8B aligned
    const int s = chunk * 256 + threadIdx.x;
    float x = (float)s * (1.0f / (float)(SEQ - 1)); // linspace(0,1,S)
    const float dt = 1.0f / (float)NSTEPS;
    #pragma unroll
    for (int it = 0; it < NSTEPS; ++it) {
        int c = (int)(x * (float)NCELLS);           // trunc toward zero, as astype(int32)
        c = c < 0 ? 0 : (c > NCELLS - 1 ? NCELLS - 1 : c);
        const float2 ab = cp[c];
        x = fmaf(fmaf(ab.x, x, ab.y), dt, x);       // x += (a*x + b)*dt
    }
    out[(size_t)b * SEQ + s] = x;
}

// ---------------------------------------------------------------------------
// Launch
// ---------------------------------------------------------------------------
extern "C" void kernel_launch(void* const* d_in, const int* in_sizes, int n_in,
                              void* d_out, int out_size, void* d_ws, size_t ws_size,
                              hipStream_t stream) {
    (void)in_sizes; (void)n_in; (void)out_size; (void)ws_size;
    const float* input_seq = (const float*)d_in[0];   // (64, 4096, 128)
    const float* W_loc     = (const float*)d_in[1];   // (128, 63)
    const float* b_loc     = (const float*)d_in[2];   // (63,)
    const float* basis     = (const float*)d_in[3];   // (128, 63)
    float*       out       = (float*)d_out;           // (64, 4096)

    // Workspace layout (floats)
    float* ws      = (float*)d_ws;
    float* partial = ws;                                    // 16*64*128 = 131072
    float* pooled  = partial + NCHUNK * BSZ * DIM;          // 8192
    float* Mmat    = pooled + BSZ * DIM;                    // 16384
    float* cvec    = Mmat + 128 * 128;                      // 128
    float* Acoef   = cvec + 128;                            // 8192

    pool_kernel  <<<BSZ * NCHUNK, 256, 0, stream>>>(input_seq, partial);
    reduce_kernel<<<(BSZ * DIM) / 256, 256, 0, stream>>>(partial, pooled);
    cvec_kernel  <<<1, 128, 0, stream>>>(b_loc, basis, cvec);
    prep_M_kernel<<<8, 256, 0, stream>>>(W_loc, basis, Mmat);
    gemm_A_kernel<<<4, 256, 0, stream>>>(pooled, Mmat, cvec, Acoef);
    ode_kernel   <<<BSZ * (SEQ / 256), 256, 0, stream>>>(Acoef, out);
}